// rbf_svm_dual_75196287418804
// MI455X (gfx1250) — compile-verified
//
#include <hip/hip_runtime.h>
#include <hip/hip_bf16.h>
#include <math.h>

typedef float v2f __attribute__((ext_vector_type(2)));
typedef float v8f __attribute__((ext_vector_type(8)));

#define GAMMA 0.1f
#define B_SZ  256
#define D_DIM 256
#define P_SZ  512
#define Q_SZ  512

// ---------------------------------------------------------------------------
// Kernel 1: per-SV negated bias:  nbias[i] = g*||S_i||^2 - log(alpha_i)
//   indices [0,512)    -> SVs_1 / alpha_1
//   indices [512,1024) -> SVs_0 / alpha_0
// ---------------------------------------------------------------------------
__global__ __launch_bounds__(256)
void sv_bias_kernel(const float* __restrict__ SVs_1,
                    const float* __restrict__ SVs_0,
                    const float* __restrict__ alpha_1,
                    const float* __restrict__ alpha_0,
                    float* __restrict__ nbias) {
    const int idx = blockIdx.x * blockDim.x + threadIdx.x;   // 0..1023
    const float* S;
    float a;
    int r;
    if (idx < P_SZ) { S = SVs_1; r = idx;         a = alpha_1[r]; }
    else            { S = SVs_0; r = idx - P_SZ;  a = alpha_0[r]; }
    const float4* row = (const float4*)(S + (size_t)r * D_DIM);
    float s = 0.0f;
#pragma unroll 8
    for (int i = 0; i < D_DIM / 4; ++i) {
        float4 v = row[i];
        s += v.x * v.x + v.y * v.y + v.z * v.z + v.w * v.w;
    }
    nbias[idx] = GAMMA * s - logf(a);
}

// ---------------------------------------------------------------------------
// Kernel 2: per 16-row batch tile:
//   vA[m, p] = -2g * (x_m . S1_p) + nbias[p]        (p = 0..511)
//   vC[m, q] = -2g * (x_m . S0_q) + nbias[512+q]    (q = 0..511)
// via V_WMMA_F32_16X16X4_F32 (8 live accumulator tiles per wave, K chunked
// by 32 so X fragments are loaded once per chunk and reused across tiles),
// then out[m] = logsumexp_q(vC[m,:]) - logsumexp_p(vA[m,:]).
// ---------------------------------------------------------------------------
__global__ __launch_bounds__(256)
void rbf_lse_kernel(const float* __restrict__ X,
                    const float* __restrict__ S1,
                    const float* __restrict__ S0,
                    const float* __restrict__ nbias,
                    float* __restrict__ out) {
    // [region (A=0,C=1)][m 0..15][n 0..511]  -> 64 KB
    __shared__ float v_lds[2 * 16 * 512];

    const int lane  = threadIdx.x & 31;
    const int wave  = threadIdx.x >> 5;          // 0..7
    const int mbase = blockIdx.x * 16;           // batch-row tile base

    const int lm = lane & 15;                    // M (or N) index within tile
    const int kh = lane >> 4;                    // K-half selector (0/1)

    // Per-lane operand bases (ISA f32 16x4 layouts):
    //  A (16x4): lanes 0-15 hold K=k0,k0+1 for M=lane; lanes 16-31 K=k0+2,k0+3.
    //  B (4x16) = S^T: lane holds column N = nbase+lm, same K-pair rule.
    const float* xp  = X  + (size_t)(mbase + lm)     * D_DIM + 2 * kh;
    const float* s1p = S1 + (size_t)(wave * 16 + lm) * D_DIM + 2 * kh;
    const float* s0p = S0 + (size_t)(wave * 16 + lm) * D_DIM + 2 * kh;

    // Wave w owns tiles t = w + 8j, j = 0..7.  j<4 -> S1 region, j>=4 -> S0.
    // Tile-to-tile stride is 128 S-rows = 128*256 floats (uniform imm offset).
    v8f acc[8];
#pragma unroll
    for (int j = 0; j < 8; ++j) acc[j] = (v8f){};

    for (int kc = 0; kc < D_DIM; kc += 32) {     // 8 chunks of K=32
        v2f xa[8];
#pragma unroll
        for (int i = 0; i < 8; ++i)
            xa[i] = *(const v2f*)(xp + kc + 4 * i);
#pragma unroll
        for (int j = 0; j < 8; ++j) {
            const float* sp = (j < 4) ? (s1p + (size_t)j * (128 * D_DIM))
                                      : (s0p + (size_t)(j - 4) * (128 * D_DIM));
#pragma unroll
            for (int i = 0; i < 8; ++i) {
                v2f b = *(const v2f*)(sp + kc + 4 * i);
                acc[j] = __builtin_amdgcn_wmma_f32_16x16x4_f32(
                    false, xa[i], false, b, (short)0, acc[j], false, false);
            }
        }
    }

    // Epilogue: scale, add bias, stash into LDS.
    // C/D layout: lanes 0-15: VGPR r -> M=r; lanes 16-31: VGPR r -> M=8+r.
#pragma unroll
    for (int j = 0; j < 8; ++j) {
        const int t     = wave + 8 * j;
        const int reg   = (t < 32) ? 0 : 1;
        const int nbase = (t & 31) * 16;
        const float nb  = nbias[reg * P_SZ + nbase + lm];
        float*     dst  = v_lds + reg * (16 * 512) + nbase + lm;
#pragma unroll
        for (int r = 0; r < 8; ++r)
            dst[(r + 8 * kh) * 512] = fmaf(-2.0f * GAMMA, acc[j][r], nb);
    }

    __syncthreads();

    // Phase 2: each wave reduces two full rows (max-subtracted logsumexp).
#pragma unroll
    for (int rr = 0; rr < 2; ++rr) {
        const int m = 2 * wave + rr;
        float lse[2];
#pragma unroll
        for (int reg = 0; reg < 2; ++reg) {
            const float* row = v_lds + reg * (16 * 512) + m * 512;
            float mx = -__builtin_inff();
#pragma unroll
            for (int i = 0; i < 16; ++i)
                mx = fmaxf(mx, row[lane + 32 * i]);
#pragma unroll
            for (int off = 16; off >= 1; off >>= 1)
                mx = fmaxf(mx, __shfl_xor(mx, off, 32));
            float s = 0.0f;
#pragma unroll
            for (int i = 0; i < 16; ++i)
                s += expf(row[lane + 32 * i] - mx);
#pragma unroll
            for (int off = 16; off >= 1; off >>= 1)
                s += __shfl_xor(s, off, 32);
            lse[reg] = mx + logf(s);
        }
        if (lane == 0)
            out[mbase + m] = lse[1] - lse[0];
    }
}

// ---------------------------------------------------------------------------
// Launch
// ---------------------------------------------------------------------------
extern "C" void kernel_launch(void* const* d_in, const int* in_sizes, int n_in,
                              void* d_out, int out_size, void* d_ws, size_t ws_size,
                              hipStream_t stream) {
    (void)in_sizes; (void)n_in; (void)out_size; (void)ws_size;
    const float* x_input = (const float*)d_in[0];  // [256, 256]
    const float* SVs_1   = (const float*)d_in[1];  // [512, 256]
    const float* SVs_0   = (const float*)d_in[2];  // [512, 256]
    const float* alpha_1 = (const float*)d_in[3];  // [512]
    const float* alpha_0 = (const float*)d_in[4];  // [512]
    float*       out     = (float*)d_out;          // [256]
    float*       nbias   = (float*)d_ws;           // 1024 floats of scratch

    sv_bias_kernel<<<4, 256, 0, stream>>>(SVs_1, SVs_0, alpha_1, alpha_0, nbias);
    rbf_lse_kernel<<<B_SZ / 16, 256, 0, stream>>>(x_input, SVs_1, SVs_0, nbias, out);
}